// MTCNN_22986664968964
// MI455X (gfx1250) — compile-verified
//
#include <hip/hip_runtime.h>
#include <stdint.h>

#define HH     768
#define WW     768
#define NPB    (HH*WW)        // 589824 cells per batch
#define BATCH  8
#define KTOP   2048
#define NBINS  8192
#define CAP    16384          // candidate buffer per batch
#define CHUNK  8192           // floats per block in scan passes
#define NCHUNK (NPB/CHUNK)    // 72
#define SORTN  4096

typedef __attribute__((ext_vector_type(2))) float v2f;
typedef __attribute__((ext_vector_type(8))) float v8f;

// ---- gfx1250 async global->LDS copy (ASYNCcnt path) ---------------------
__device__ __forceinline__ void async_load_b128_to_lds(unsigned lds_byte_addr,
                                                       const void* gaddr) {
  // copies 16 bytes from global memory into LDS, tracked by ASYNCcnt
  asm volatile("global_load_async_to_lds_b128 %0, %1, off"
               :: "v"(lds_byte_addr), "v"(gaddr) : "memory");
}
__device__ __forceinline__ void wait_asynccnt0() {
  asm volatile("s_wait_asynccnt 0" ::: "memory");
}
// Low 32 bits of a generic pointer to LDS are the wave-relative LDS byte addr.
__device__ __forceinline__ unsigned lds_addr_of(const void* p) {
  return (unsigned)(uintptr_t)p;
}

// ---- Pass 1: per-batch 8192-bin histogram of score mantissa bits --------
// All scores in [0.6,1) share sign+exponent, so bits[22:10] order == value order.
__global__ __launch_bounds__(256) void k_hist(const float* __restrict__ probs,
                                              unsigned* __restrict__ ghist) {
  __shared__ float    buf[CHUNK];   // 32 KB, filled by async-to-LDS
  __shared__ unsigned lh[NBINS];    // 32 KB local histogram
  const int t = threadIdx.x;
  const int chunk = blockIdx.x, b = blockIdx.y;
  const float* src = probs + (size_t)b * NPB + (size_t)chunk * CHUNK;

  for (int i = t; i < NBINS; i += 256) lh[i] = 0u;

  // 2048 async b128 transfers: 8 per thread, contiguous 32 KB tile
#pragma unroll
  for (int i = 0; i < 8; ++i) {
    const int e = (i * 256 + t) * 4;               // float index, 16B aligned
    async_load_b128_to_lds(lds_addr_of(&buf[e]), (const void*)(src + e));
  }
  wait_asynccnt0();      // each wave waits for its own async transfers
  __syncthreads();       // make all waves' LDS writes visible

#pragma unroll 4
  for (int i = 0; i < 32; ++i) {
    const float v = buf[t * 32 + i];
    if (v >= 0.6f) {
      const unsigned bin = (__float_as_uint(v) >> 10) & (NBINS - 1);
      atomicAdd(&lh[bin], 1u);
    }
  }
  __syncthreads();

  unsigned* gh = ghist + (size_t)b * NBINS;
  for (int i = t; i < NBINS; i += 256) {
    const unsigned c = lh[i];
    if (c) atomicAdd(&gh[i], c);
  }
}

// ---- Pass 2: find threshold bin T of the K-th largest score -------------
// Uses V_WMMA_F32_16X16X4_F32 as an exact f32 reduction: A=ones(16x4),
// B=64 histogram counts (4x16) -> D row gives column sums; summing the 16
// column sums over lanes 0..15 yields the exact 64-bin block total
// (counts < 2^24, so all matrix-unit adds are exact regardless of B layout).
__global__ __launch_bounds__(32) void k_thresh(const unsigned* __restrict__ ghist,
                                               unsigned* __restrict__ thr) {
  __shared__ float fh[NBINS];       // 32 KB
  __shared__ float bsum[128];
  const int l = threadIdx.x;        // one full wave, EXEC all ones for WMMA
  const int b = blockIdx.x;
  const unsigned* gh = ghist + (size_t)b * NBINS;

  for (int i = l; i < NBINS; i += 32) fh[i] = (float)gh[i];
  __syncthreads();

  const int n    = l & 15;          // B column for this lane
  const int kblk = (l >> 4) * 2;    // B rows {kblk, kblk+1} for this lane
  v2f a; a.x = 1.0f; a.y = 1.0f;    // A = all ones (layout invariant)

  for (int g = 0; g < 128; ++g) {
    const float* base = &fh[g * 64];
    v2f bv;
    bv.x = base[n * 4 + kblk + 0];
    bv.y = base[n * 4 + kblk + 1];
    v8f c = {};
    v8f d = __builtin_amdgcn_wmma_f32_16x16x4_f32(
        /*neg_a=*/false, a, /*neg_b=*/false, bv,
        /*c_mod=*/(short)0, c, /*reuse_a=*/false, /*reuse_b=*/false);
    float v = d[0];                 // column sum for N = lane (rows identical)
    v += __shfl_xor(v, 1, 16);
    v += __shfl_xor(v, 2, 16);
    v += __shfl_xor(v, 4, 16);
    v += __shfl_xor(v, 8, 16);      // lanes 0..15 -> exact 64-bin block total
    if (l == 0) bsum[g] = v;
  }
  __syncthreads();

  if (l == 0) {
    float cum = 0.0f;
    int T = 0;
    for (int g = 127; g >= 0; --g) {
      if (cum + bsum[g] >= (float)KTOP) {       // K-th element lies in block g
        float c2 = cum;
        int bin = g * 64 + 63;
        for (; bin >= g * 64; --bin) {
          c2 += fh[bin];
          if (c2 >= (float)KTOP) break;
        }
        T = (bin < g * 64) ? g * 64 : bin;
        break;
      }
      cum += bsum[g];
    }
    thr[b] = (unsigned)T;   // T==0 => fewer than K candidates, collect all
  }
}

// ---- Pass 3: collect every candidate with bin >= T ----------------------
// Key = (scoreBits << 32) | ~idx : descending sort == top_k order
// (value desc, index asc on ties).
__global__ __launch_bounds__(256) void k_collect(const float* __restrict__ probs,
                                                 const unsigned* __restrict__ thr,
                                                 unsigned* __restrict__ cnt,
                                                 unsigned long long* __restrict__ cand) {
  const int t = threadIdx.x, chunk = blockIdx.x, b = blockIdx.y;
  const unsigned T = thr[b];
  const float* src = probs + (size_t)b * NPB + (size_t)chunk * CHUNK;
  unsigned long long* cb = cand + (size_t)b * CAP;

#pragma unroll
  for (int i = 0; i < 8; ++i) {
    const int e = (i * 256 + t) * 4;
    const float4 v4 = *(const float4*)(src + e);
    const float vv[4] = {v4.x, v4.y, v4.z, v4.w};
#pragma unroll
    for (int c = 0; c < 4; ++c) {
      const float v = vv[c];
      if (v >= 0.6f) {
        const unsigned bits = __float_as_uint(v);
        if (((bits >> 10) & (NBINS - 1)) >= T) {
          const unsigned pos = atomicAdd(&cnt[b], 1u);
          if (pos < CAP) {
            const unsigned idx = (unsigned)(chunk * CHUNK + e + c);
            cb[pos] = ((unsigned long long)bits << 32) |
                      (unsigned long long)(unsigned)(~idx);
          }
        }
      }
    }
  }
}

// ---- Pass 4: bitonic sort of <=4096 survivors, emit exact top-K ---------
__global__ __launch_bounds__(1024) void k_sort(const unsigned* __restrict__ cnt,
                                               const unsigned long long* __restrict__ cand,
                                               unsigned long long* __restrict__ topk) {
  __shared__ unsigned long long s[SORTN];   // 32 KB
  const int t = threadIdx.x, b = blockIdx.x;
  unsigned m = cnt[b];
  if (m > CAP) m = CAP;
  if (m > SORTN) m = SORTN;                 // expected m ~ 2.2K << 4096
  const unsigned long long* cb = cand + (size_t)b * CAP;
  for (int i = t; i < SORTN; i += 1024) s[i] = (i < (int)m) ? cb[i] : 0ull;
  __syncthreads();

  for (int k = 2; k <= SORTN; k <<= 1) {
    for (int j = k >> 1; j > 0; j >>= 1) {
      for (int i = t; i < SORTN; i += 1024) {
        const int ixj = i ^ j;
        if (ixj > i) {
          const unsigned long long A = s[i], B = s[ixj];
          const bool desc = ((i & k) == 0);
          if (desc ? (A < B) : (A > B)) { s[i] = B; s[ixj] = A; }
        }
      }
      __syncthreads();
    }
  }
  unsigned long long* tb = topk + (size_t)b * KTOP;
  for (int i = t; i < KTOP; i += 1024) tb[i] = s[i];
}

// ---- Pass 5: greedy NMS (IOU>=0.5 == Chebyshev<=1 on the stride-4 grid,
//              exact for 23x23 boxes) + refine + square + clip + write ----
__global__ __launch_bounds__(256) void k_nms(const unsigned long long* __restrict__ topk,
                                             const float* __restrict__ devs,
                                             float* __restrict__ out) {
  __shared__ unsigned map[HH * (WW / 32)];  // 72 KB kept-cell bitmap
  __shared__ unsigned char keep[KTOP];
  const int t = threadIdx.x, b = blockIdx.x;
  const unsigned long long* tb = topk + (size_t)b * KTOP;

  for (int i = t; i < HH * (WW / 32); i += 256) map[i] = 0u;
  __syncthreads();

  if (t == 0) {  // greedy sweep is inherently sequential (rank order)
    for (int r = 0; r < KTOP; ++r) {
      const unsigned long long key = tb[r];
      const float sc = __uint_as_float((unsigned)(key >> 32));
      unsigned kp = 0;
      if (sc >= 0.6f) {
        const unsigned idx = ~(unsigned)(key & 0xFFFFFFFFu);
        const int y = (int)(idx / WW), x = (int)(idx - (unsigned)y * WW);
        const int y0 = (y > 0) ? y - 1 : 0, y1 = (y < HH - 1) ? y + 1 : HH - 1;
        const int x0 = (x > 0) ? x - 1 : 0, x1 = (x < WW - 1) ? x + 1 : WW - 1;
        unsigned hit = 0;
        for (int yy = y0; yy <= y1; ++yy) {
          const int rb = yy * (WW / 32);
          for (int xx = x0; xx <= x1; ++xx)
            hit |= (map[rb + (xx >> 5)] >> (xx & 31)) & 1u;
        }
        if (!hit) { kp = 1; map[y * (WW / 32) + (x >> 5)] |= 1u << (x & 31); }
      }
      keep[r] = (unsigned char)kp;
    }
  }
  __syncthreads();

  for (int r = t; r < KTOP; r += 256) {
    float* o = out + ((size_t)b * KTOP + r) * 5;
    const unsigned long long key = tb[r];
    const float sc = __uint_as_float((unsigned)(key >> 32));
    if (!keep[r]) {  // suppressed or invalid: reference emits clip(...)*0 == +0
      o[0] = 0.0f; o[1] = 0.0f; o[2] = 0.0f; o[3] = 0.0f; o[4] = 0.0f;
      continue;
    }
    const unsigned idx = ~(unsigned)(key & 0xFFFFFFFFu);
    const int y = (int)(idx / WW), x = (int)(idx % WW);
    const float uly = (float)(4 * y + 2),  ulx = (float)(4 * x + 2);
    const float dry = (float)(4 * y + 24), drx = (float)(4 * x + 24);
    const float* dv = devs + ((size_t)b * NPB + idx) * 4;
    const float r0 = uly + dv[0] * 22.0f, r1 = ulx + dv[1] * 22.0f;
    const float r2 = dry + dv[2] * 22.0f, r3 = drx + dv[3] * 22.0f;
    const float rh = r2 - r0, rw = r3 - r1;
    const float len = fmaxf(rh, rw);
    float suly = (r0 + rh * 0.5f) - 0.5f * len;
    float sulx = (r1 + rw * 0.5f) - 0.5f * len;
    float sdry = suly + len, sdrx = sulx + len;
    const float lo = 1.0f, hi = 3095.0f;   // ceil((768*2+12)/0.5) - 1
    suly = fminf(fmaxf(suly, lo), hi);
    sulx = fminf(fmaxf(sulx, lo), hi);
    sdry = fminf(fmaxf(sdry, lo), hi);
    sdrx = fminf(fmaxf(sdrx, lo), hi);
    o[0] = suly; o[1] = sulx; o[2] = sdry; o[3] = sdrx; o[4] = sc;
  }
}

extern "C" void kernel_launch(void* const* d_in, const int* in_sizes, int n_in,
                              void* d_out, int out_size, void* d_ws, size_t ws_size,
                              hipStream_t stream) {
  (void)in_sizes; (void)n_in; (void)out_size; (void)ws_size;
  const float* probs = (const float*)d_in[0];
  const float* devs  = (const float*)d_in[1];
  char* ws = (char*)d_ws;
  // workspace layout (~1.44 MB total)
  unsigned* hist = (unsigned*)ws;                          // 262144 B
  unsigned* cnt  = (unsigned*)(ws + 262144);               //     32 B
  unsigned* thr  = (unsigned*)(ws + 262144 + 32);          //     32 B
  unsigned long long* cand = (unsigned long long*)(ws + 262400);           // 1 MB
  unsigned long long* topk = (unsigned long long*)(ws + 262400 +
                                                   (size_t)BATCH * CAP * 8); // 128 KB

  hipMemsetAsync(ws, 0, 262144 + 32, stream);  // zero hist + cnt each launch

  k_hist   <<<dim3(NCHUNK, BATCH), 256,  0, stream>>>(probs, hist);
  k_thresh <<<dim3(BATCH),          32,  0, stream>>>(hist, thr);
  k_collect<<<dim3(NCHUNK, BATCH), 256,  0, stream>>>(probs, thr, cnt, cand);
  k_sort   <<<dim3(BATCH),        1024,  0, stream>>>(cnt, cand, topk);
  k_nms    <<<dim3(BATCH),         256,  0, stream>>>(topk, devs, (float*)d_out);
}